// GatedOpponentCSSM_49125835931748
// MI455X (gfx1250) — compile-verified
//
#include <hip/hip_runtime.h>
#include <math.h>

// ---------------------------------------------------------------------------
// GatedOpponentCSSM on MI455X (gfx1250, wave32).
//   - rfft2 / irfft2 realized as DFT-matrix matmuls via V_WMMA_F32_16X16X4_F32
//   - bulk global->LDS staging via GLOBAL_LOAD_ASYNC_TO_LDS_B128 (ASYNCcnt)
//   - per-bin sequential GOOM scan over T (262144 independent lanes)
//   - gate GEMMs (64x32 @ 32x32) also via WMMA f32
// ---------------------------------------------------------------------------

typedef float v2f __attribute__((ext_vector_type(2)));
typedef float v8f __attribute__((ext_vector_type(8)));

#define LDP 65  // padded LDS row stride for 64-wide planes (64 banks -> conflict-free)

__device__ __forceinline__ int lane_id() { return (int)(threadIdx.x & 31); }

// Async copy of 16 contiguous bytes global -> LDS (CDNA5 async-load unit,
// tracked by ASYNCcnt; LDS written directly, no VGPR round-trip).
__device__ __forceinline__ void async_cp16(const float* __restrict__ g, float* l) {
  unsigned long long ga = (unsigned long long)(uintptr_t)g;
  unsigned int la = (unsigned int)(uintptr_t)l;  // low 32 bits = LDS byte offset
  asm volatile("global_load_async_to_lds_b128 %0, %1, off" ::"v"(la), "v"(ga)
               : "memory");
}
__device__ __forceinline__ void async_wait0() {
  asm volatile("s_wait_asynccnt 0x0" ::: "memory");
}

// D(16x16) += A(16xK) @ B(Kx16). A,B live in LDS with row strides lda/ldb.
// f32 operand layout (ISA 7.12.2, 32-bit A 16x4): lanes 0-15 hold K=0/1,
// lanes 16-31 hold K=2/3 -> per lane 2 consecutive K values at (lane>>4)*2.
__device__ __forceinline__ v8f wmma_mac_f32(const float* __restrict__ As, int lda, int m0,
                                            const float* __restrict__ Bs, int ldb, int n0,
                                            int K, v8f acc) {
  const int l = lane_id();
  const int m = m0 + (l & 15);
  const int n = n0 + (l & 15);
  const int kk = (l >> 4) * 2;
  for (int k0 = 0; k0 < K; k0 += 4) {
    v2f a, b;
    a.x = As[m * lda + (k0 + kk)];
    a.y = As[m * lda + (k0 + kk + 1)];
    b.x = Bs[(k0 + kk) * ldb + n];
    b.y = Bs[(k0 + kk + 1) * ldb + n];
    acc = __builtin_amdgcn_wmma_f32_16x16x4_f32(false, a, false, b, (short)0, acc,
                                                false, false);
  }
  return acc;
}

// C/D layout (ISA 7.12.2): VGPR r -> M=r (lanes 0-15), M=r+8 (lanes 16-31), N=lane%16.
__device__ __forceinline__ void store_tile_lds(float* D, int m0, int n0, v8f acc) {
  const int l = lane_id();
  const int n = n0 + (l & 15);
  const int mb = m0 + ((l >> 4) << 3);
#pragma unroll
  for (int r = 0; r < 8; ++r) D[(mb + r) * LDP + n] = acc[r];
}

// ---------------------------------------------------------------------------
// Basis: F[w][k] = exp(-2*pi*i*w*k/64), symmetric 64x64.
__global__ __launch_bounds__(256) void basis_kernel(float* __restrict__ Fre,
                                                    float* __restrict__ Fim) {
  int tid = blockIdx.x * 256 + threadIdx.x;
  if (tid < 4096) {
    int w = tid >> 6, k = tid & 63;
    float ang = -2.0f * 3.14159265358979323846f * (float)((w * k) & 63) / 64.0f;
    float s, c;
    sincosf(ang, &s, &c);
    Fre[tid] = c;
    Fim[tid] = s;
  }
}

// Pad k_exc (planes 0..31) / k_inh (planes 32..63) 15x15 -> 64x64 at offset (24,24).
__global__ __launch_bounds__(256) void pad_kernel(const float* __restrict__ kexc,
                                                  const float* __restrict__ kinh,
                                                  float* __restrict__ pad) {
  int p = blockIdx.x;
  float* dst = pad + (size_t)p * 4096;
  const float* src = (p < 32) ? (kexc + p * 225) : (kinh + (p - 32) * 225);
  for (int i = threadIdx.x; i < 4096; i += 256) {
    int h = i >> 6, w = i & 63;
    float v = 0.0f;
    if (h >= 24 && h < 39 && w >= 24 && w < 39) v = src[(h - 24) * 15 + (w - 24)];
    dst[i] = v;
  }
}

// ---------------------------------------------------------------------------
// Forward fft2 of one 64x64 real plane via two DFT matmul passes (full spectrum;
// Hermitian symmetry is preserved through the per-bin scan, so irfft == Re(ifft2)).
// src_is_x=1: src is (B,T,H,W,C) channel-last, plane p = bt*32+c (L2 absorbs the
// stride-32 gathers; 192 MB L2 >> 512 KB of lines per (b,t)).
__global__ __launch_bounds__(256) void fft2_fwd_kernel(const float* __restrict__ src,
                                                       int src_is_x,
                                                       const float* __restrict__ Fre_g,
                                                       const float* __restrict__ Fim_g,
                                                       float* __restrict__ Ure,
                                                       float* __restrict__ Uim) {
  __shared__ float lds[5 * 64 * LDP];
  float* fre = lds;
  float* fim = lds + 64 * LDP;
  float* xs  = lds + 2 * 64 * LDP;
  float* tre = lds + 3 * 64 * LDP;
  float* tim = lds + 4 * 64 * LDP;
  const int p = blockIdx.x;
  const int tid = threadIdx.x;
  // Async-stage the DFT basis (row-contiguous 16B chunks).
  for (int idx = tid; idx < 1024; idx += 256) {
    int h = idx >> 4, wc = (idx & 15) << 2;
    async_cp16(Fre_g + h * 64 + wc, &fre[h * LDP + wc]);
    async_cp16(Fim_g + h * 64 + wc, &fim[h * LDP + wc]);
  }
  if (src_is_x) {
    // channel-last gather (stride 32): synchronous path.
    int bt = p >> 5, c = p & 31;
    for (int i = tid; i < 4096; i += 256) {
      int h = i >> 6, w = i & 63;
      xs[h * LDP + w] = src[((size_t)bt * 4096 + i) * 32 + c];
    }
  } else {
    const float* sp = src + (size_t)p * 4096;
    for (int idx = tid; idx < 1024; idx += 256) {
      int h = idx >> 4, wc = (idx & 15) << 2;
      async_cp16(sp + h * 64 + wc, &xs[h * LDP + wc]);
    }
  }
  async_wait0();
  __syncthreads();
  const int wv = tid >> 5;  // wave id 0..7 (wave32)
  // Stage W: T[h][kw] = sum_w x[h][w] * F[w][kw]
  for (int tt = wv; tt < 16; tt += 8) {
    int m0 = (tt >> 2) * 16, n0 = (tt & 3) * 16;
    v8f ar = {}, ai = {};
    ar = wmma_mac_f32(xs, LDP, m0, fre, LDP, n0, 64, ar);
    ai = wmma_mac_f32(xs, LDP, m0, fim, LDP, n0, 64, ai);
    store_tile_lds(tre, m0, n0, ar);
    store_tile_lds(tim, m0, n0, ai);
  }
  __syncthreads();
  // Stage H: U[kh][kw] = sum_h F[kh][h] * T[h][kw]   (F symmetric)
  const size_t pb = (size_t)p * 4096;
  for (int tt = wv; tt < 16; tt += 8) {
    int m0 = (tt >> 2) * 16, n0 = (tt & 3) * 16;
    v8f rr = {}, ii = {}, ri = {}, ir = {};
    rr = wmma_mac_f32(fre, LDP, m0, tre, LDP, n0, 64, rr);
    ii = wmma_mac_f32(fim, LDP, m0, tim, LDP, n0, 64, ii);
    ri = wmma_mac_f32(fre, LDP, m0, tim, LDP, n0, 64, ri);
    ir = wmma_mac_f32(fim, LDP, m0, tre, LDP, n0, 64, ir);
    v8f ure = rr - ii;
    v8f uim = ri + ir;
    const int l = lane_id();
    const int n = n0 + (l & 15);
    const int mb = m0 + ((l >> 4) << 3);
#pragma unroll
    for (int r = 0; r < 8; ++r) {
      Ure[pb + (size_t)(mb + r) * 64 + n] = ure[r];
      Uim[pb + (size_t)(mb + r) * 64 + n] = uim[r];
    }
  }
}

// Inverse: y = Re(ifft2(Y)) / 4096, written back over Ure plane.
__global__ __launch_bounds__(256) void fft2_inv_kernel(const float* __restrict__ Fre_g,
                                                       const float* __restrict__ Fim_g,
                                                       float* __restrict__ Ure,
                                                       float* __restrict__ Uim) {
  __shared__ float lds[6 * 64 * LDP];
  float* fre = lds;
  float* fim = lds + 64 * LDP;
  float* yre = lds + 2 * 64 * LDP;
  float* yim = lds + 3 * 64 * LDP;
  float* zre = lds + 4 * 64 * LDP;
  float* zim = lds + 5 * 64 * LDP;
  const int p = blockIdx.x;
  const int tid = threadIdx.x;
  const size_t pb = (size_t)p * 4096;
  for (int idx = tid; idx < 1024; idx += 256) {
    int h = idx >> 4, wc = (idx & 15) << 2;
    size_t go = (size_t)(h * 64 + wc);
    async_cp16(Fre_g + go, &fre[h * LDP + wc]);
    async_cp16(Fim_g + go, &fim[h * LDP + wc]);
    async_cp16(Ure + pb + go, &yre[h * LDP + wc]);
    async_cp16(Uim + pb + go, &yim[h * LDP + wc]);
  }
  async_wait0();
  __syncthreads();
  const int wv = tid >> 5;
  // Stage H: Z = conj(F) @ Y : Zre = Fre@Yre + Fim@Yim ; Zim = Fre@Yim - Fim@Yre
  for (int tt = wv; tt < 16; tt += 8) {
    int m0 = (tt >> 2) * 16, n0 = (tt & 3) * 16;
    v8f a = {}, b = {}, c = {}, d = {};
    a = wmma_mac_f32(fre, LDP, m0, yre, LDP, n0, 64, a);
    b = wmma_mac_f32(fim, LDP, m0, yim, LDP, n0, 64, b);
    c = wmma_mac_f32(fre, LDP, m0, yim, LDP, n0, 64, c);
    d = wmma_mac_f32(fim, LDP, m0, yre, LDP, n0, 64, d);
    store_tile_lds(zre, m0, n0, a + b);
    store_tile_lds(zim, m0, n0, c - d);
  }
  __syncthreads();
  // Stage W: y = Re(Z @ conj(F)) = Zre@Fre + Zim@Fim ; scale 1/4096
  for (int tt = wv; tt < 16; tt += 8) {
    int m0 = (tt >> 2) * 16, n0 = (tt & 3) * 16;
    v8f a = {};
    a = wmma_mac_f32(zre, LDP, m0, fre, LDP, n0, 64, a);
    a = wmma_mac_f32(zim, LDP, m0, fim, LDP, n0, 64, a);
    const int l = lane_id();
    const int n = n0 + (l & 15);
    const int mb = m0 + ((l >> 4) << 3);
#pragma unroll
    for (int r = 0; r < 8; ++r)
      Ure[pb + (size_t)(mb + r) * 64 + n] = a[r] * (1.0f / 4096.0f);
  }
}

// ---------------------------------------------------------------------------
// Context means: one block per (b,t); reduce 4096 spatial positions x 32 channels.
__global__ __launch_bounds__(256) void ctx_kernel(const float* __restrict__ x,
                                                  float* __restrict__ ctx) {
  __shared__ float red[256];
  const int bt = blockIdx.x;
  const int tid = threadIdx.x;
  const int c = tid & 31, j = tid >> 5;
  const float* base = x + (size_t)bt * 4096 * 32;
  float acc = 0.0f;
  for (int s = j; s < 4096; s += 8) acc += base[s * 32 + c];
  red[tid] = acc;
  __syncthreads();
  if (tid < 32) {
    float s = 0.0f;
#pragma unroll
    for (int k = 0; k < 8; ++k) s += red[k * 32 + tid];
    ctx[bt * 32 + tid] = s * (1.0f / 4096.0f);
  }
}

// Gate GEMMs: (64x32) @ (32x32) + bias, 4 gates, WMMA f32 16x16x4, one block.
__global__ __launch_bounds__(256) void gates_kernel(
    const float* __restrict__ ctx, const float* __restrict__ wA,
    const float* __restrict__ bA, const float* __restrict__ wD,
    const float* __restrict__ bD, const float* __restrict__ wM,
    const float* __restrict__ bM, const float* __restrict__ wG,
    const float* __restrict__ bG, float* __restrict__ out /* 4 x 2048 */) {
  __shared__ float sctx[64 * 33];
  __shared__ float sw[4][32 * 33];
  const int tid = threadIdx.x;
  for (int i = tid; i < 2048; i += 256) sctx[(i >> 5) * 33 + (i & 31)] = ctx[i];
  for (int i = tid; i < 1024; i += 256) {
    int r = i >> 5, cn = i & 31;
    sw[0][r * 33 + cn] = wA[i];
    sw[1][r * 33 + cn] = wD[i];
    sw[2][r * 33 + cn] = wM[i];
    sw[3][r * 33 + cn] = wG[i];
  }
  __syncthreads();
  const float* bias[4] = {bA, bD, bM, bG};
  const int wv = tid >> 5;
  for (int tt = wv; tt < 32; tt += 8) {
    int g = tt >> 3, sub = tt & 7;
    int m0 = (sub >> 1) * 16, n0 = (sub & 1) * 16;
    v8f acc = {};
    acc = wmma_mac_f32(sctx, 33, m0, sw[g], 33, n0, 32, acc);
    const int l = lane_id();
    const int n = n0 + (l & 15);
    const int mb = m0 + ((l >> 4) << 3);
    float bv = bias[g][n];
#pragma unroll
    for (int r = 0; r < 8; ++r) {
      float v = acc[r] + bv;
      float o;
      if (g < 2) o = 1.0f / (1.0f + __expf(-v));             // sigmoid
      else o = (v > 20.0f) ? v : log1pf(__expf(v));          // softplus
      out[g * 2048 + (mb + r) * 32 + n] = o;
    }
  }
}

// ---------------------------------------------------------------------------
// GOOM scan: each lane owns one (b,c,kh,kw) bin and runs the T=32 recurrence
// s_t = A_t s_{t-1} + u_t in complex log space; writes Y = exp(s_t[1]) in place.
__device__ __forceinline__ void lse3(float ar, float ai, float br, float bi,
                                     float cr, float ci, float* orr, float* oi) {
  float m = fmaxf(ar, fmaxf(br, cr));
  float ea = __expf(ar - m), eb = __expf(br - m), ec = __expf(cr - m);
  float sa, ca, sb, cb, sc, cc;
  __sincosf(ai, &sa, &ca);
  __sincosf(bi, &sb, &cb);
  __sincosf(ci, &sc, &cc);
  float zr = ea * ca + eb * cb + ec * cc;
  float zi = ea * sa + eb * sb + ec * sc;
  *orr = m + 0.5f * __logf(zr * zr + zi * zi + 1e-45f);
  *oi = atan2f(zi, zr);
}

__global__ __launch_bounds__(256) void scan_kernel(
    const float* __restrict__ gates /* alpha,delta,mu,gamma: 4 x (B*T*C) */,
    const float* __restrict__ decay, const float* __restrict__ SPre,
    const float* __restrict__ SPim, float* __restrict__ Ure,
    float* __restrict__ Uim) {
  const size_t gid = (size_t)blockIdx.x * 256 + threadIdx.x;  // B*C*4096
  const int pos = (int)(gid & 4095);
  const int c = (int)((gid >> 12) & 31);
  const int b = (int)(gid >> 17);
  float dec = fminf(fmaxf(decay[c], 0.1f), 0.99f);
  const float ldec = __logf(dec);
  // Log(K_E) and Log(-K_I) (time-invariant per bin)
  float ker = SPre[c * 4096 + pos], kei = SPim[c * 4096 + pos];
  float kir = SPre[(32 + c) * 4096 + pos], kii = SPim[(32 + c) * 4096 + pos];
  const float lE_r = 0.5f * __logf(ker * ker + kei * kei + 1e-45f);
  const float lE_i = atan2f(kei, ker);
  const float lI_r = 0.5f * __logf(kir * kir + kii * kii + 1e-45f);
  const float lI_i = atan2f(-kii, -kir);
  const float LU1 = -23.025850929940457f;  // log(1e-10)
  float s0r = 0.f, s0i = 0.f, s1r = 0.f, s1i = 0.f;
  for (int t = 0; t < 32; ++t) {
    const int bt = b * 32 + t;
    const size_t up = ((size_t)bt * 32 + c) * 4096 + pos;
    float ur = Ure[up], ui = Uim[up];
    float u0r = 0.5f * __logf(ur * ur + ui * ui + 1e-45f);
    float u0i = atan2f(ui, ur);
    if (t == 0) {
      s0r = u0r; s0i = u0i; s1r = LU1; s1i = 0.0f;
    } else {
      const float la = __logf(gates[bt * 32 + c]);
      const float ld = __logf(gates[2048 + bt * 32 + c]);
      const float lm = __logf(gates[4096 + bt * 32 + c] + 1e-45f);
      const float lg = __logf(gates[6144 + bt * 32 + c] + 1e-45f);
      // K rows: [dec*alpha, -K_I*mu ; K_E*gamma, dec*delta]
      float n0r, n0i, n1r, n1i;
      lse3(ldec + la + s0r, s0i, lI_r + lm + s1r, lI_i + s1i, u0r, u0i, &n0r, &n0i);
      lse3(lE_r + lg + s0r, lE_i + s0i, ldec + ld + s1r, s1i, LU1, 0.0f, &n1r, &n1i);
      s0r = n0r; s0i = n0i; s1r = n1r; s1i = n1i;
    }
    float er = __expf(s1r), sn, cs;
    __sincosf(s1i, &sn, &cs);
    Ure[up] = er * cs;  // Y_hat re (in place)
    Uim[up] = er * sn;  // Y_hat im
  }
}

// ---------------------------------------------------------------------------
// (B,T,C,H,W) planes -> (B,T,H,W,C) channel-last, fully coalesced via LDS tile
// (async-staged reads; stride-66 tile keeps the transposed reads conflict-free).
__global__ __launch_bounds__(256) void transpose_out_kernel(
    const float* __restrict__ yplanar, float* __restrict__ out) {
  __shared__ float t[32 * 66];
  const int blk = blockIdx.x;  // B*T*H
  const int bt = blk >> 6, h = blk & 63;
  const int tid = threadIdx.x;
  for (int idx = tid; idx < 512; idx += 256) {
    int c = idx >> 4, wc = (idx & 15) << 2;
    async_cp16(yplanar + ((size_t)bt * 32 + c) * 4096 + h * 64 + wc,
               &t[c * 66 + wc]);
  }
  async_wait0();
  __syncthreads();
  float* ob = out + ((size_t)bt * 64 + h) * 64 * 32;
  for (int i = tid; i < 2048; i += 256) {
    int w = i >> 5, c = i & 31;
    ob[i] = t[c * 66 + w];
  }
}

// ---------------------------------------------------------------------------
extern "C" void kernel_launch(void* const* d_in, const int* in_sizes, int n_in,
                              void* d_out, int out_size, void* d_ws, size_t ws_size,
                              hipStream_t stream) {
  (void)in_sizes; (void)n_in; (void)out_size; (void)ws_size;
  const float* x    = (const float*)d_in[0];
  const float* wA   = (const float*)d_in[1];
  const float* bA   = (const float*)d_in[2];
  const float* wD   = (const float*)d_in[3];
  const float* bD   = (const float*)d_in[4];
  const float* wM   = (const float*)d_in[5];
  const float* bM   = (const float*)d_in[6];
  const float* wG   = (const float*)d_in[7];
  const float* bG   = (const float*)d_in[8];
  const float* dec  = (const float*)d_in[9];
  const float* kexc = (const float*)d_in[10];
  const float* kinh = (const float*)d_in[11];

  float* ws = (float*)d_ws;
  const size_t OFF_FRE   = 0;                         // 4096
  const size_t OFF_FIM   = OFF_FRE + 4096;            // 4096
  const size_t OFF_CTX   = OFF_FIM + 4096;            // 2048
  const size_t OFF_GATES = OFF_CTX + 2048;            // 4*2048
  const size_t OFF_PAD   = OFF_GATES + 8192;          // 64*4096
  const size_t OFF_SPRE  = OFF_PAD + 262144;          // 64*4096
  const size_t OFF_SPIM  = OFF_SPRE + 262144;         // 64*4096
  const size_t OFF_URE   = OFF_SPIM + 262144;         // 2048*4096
  const size_t OFF_UIM   = OFF_URE + 8388608;         // 2048*4096
  float* Fre = ws + OFF_FRE;
  float* Fim = ws + OFF_FIM;
  float* ctx = ws + OFF_CTX;
  float* gts = ws + OFF_GATES;
  float* pad = ws + OFF_PAD;
  float* SPr = ws + OFF_SPRE;
  float* SPi = ws + OFF_SPIM;
  float* Ure = ws + OFF_URE;
  float* Uim = ws + OFF_UIM;

  basis_kernel<<<16, 256, 0, stream>>>(Fre, Fim);
  pad_kernel<<<64, 256, 0, stream>>>(kexc, kinh, pad);
  fft2_fwd_kernel<<<64, 256, 0, stream>>>(pad, 0, Fre, Fim, SPr, SPi);
  ctx_kernel<<<64, 256, 0, stream>>>(x, ctx);
  gates_kernel<<<1, 256, 0, stream>>>(ctx, wA, bA, wD, bD, wM, bM, wG, bG, gts);
  fft2_fwd_kernel<<<2048, 256, 0, stream>>>(x, 1, Fre, Fim, Ure, Uim);
  scan_kernel<<<1024, 256, 0, stream>>>(gts, dec, SPr, SPi, Ure, Uim);
  fft2_inv_kernel<<<2048, 256, 0, stream>>>(Fre, Fim, Ure, Uim);
  transpose_out_kernel<<<4096, 256, 0, stream>>>(Ure, (float*)d_out);
}